// Conv3d_43817256354329
// MI455X (gfx1250) — compile-verified
//
#include <hip/hip_runtime.h>

typedef __attribute__((ext_vector_type(2))) float v2f;
typedef __attribute__((ext_vector_type(8))) float v8f;

#define KVOL 27
#define CIN  64
#define COUT 64
#define LDS_STRIDE 68            // 64 + 4 pad: conflict-free strided b64 reads
#define WAVES_PER_BLOCK 8
#define TILE_M 16

__global__ __launch_bounds__(256) void zero_f32_kernel(float* __restrict__ p, int n) {
    int tid = blockIdx.x * blockDim.x + threadIdx.x;
    int stride = gridDim.x * blockDim.x;
    int n4 = n >> 2;
    float4 z = make_float4(0.f, 0.f, 0.f, 0.f);
    for (int i = tid; i < n4; i += stride)
        ((float4*)p)[i] = z;
    for (int i = (n4 << 2) + tid; i < n; i += stride)
        p[i] = 0.f;
}

__global__ __launch_bounds__(256) void spconv_wmma_f32_kernel(
    const float* __restrict__ feats,    // [N, CIN]
    const float* __restrict__ weights,  // [KVOL, CIN, COUT]
    const int*   __restrict__ in_map,   // [KVOL, M]
    const int*   __restrict__ out_map,  // [KVOL, M]
    float*       __restrict__ out,      // [N, COUT]
    int M)
{
    __shared__ __align__(16) float sW[COUT * LDS_STRIDE];   // W_k transposed: sW[cout][cin]

    const int k    = blockIdx.y;
    const int tid  = threadIdx.x;
    const int wave = tid >> 5;
    const int lane = tid & 31;

    // Stage W_k (64x64 f32 = 16 KB) into LDS, transposed so one lane's two
    // consecutive-cin B values are a single 8-byte LDS load.
    const float* Wg = weights + (size_t)k * (CIN * COUT);
    for (int i = tid; i < CIN * COUT; i += 256) {
        int cin = i >> 6, cout = i & 63;
        sW[cout * LDS_STRIDE + cin] = Wg[i];
    }
    __syncthreads();

    const int tile = blockIdx.x * WAVES_PER_BLOCK + wave;
    const int m0   = tile * TILE_M;
    if (m0 + TILE_M > M) return;          // M = 50000 is a multiple of 16

    const int r     = lane & 15;          // A: row in tile; B/D: column in 16-block
    const int khalf = lane >> 4;          // which K-pair of the 4-wide slab
    const int klo   = khalf * 2;

    const size_t kbase = (size_t)k * (size_t)M;

    // Gather: lane's source row for the A fragment
    const int in_row = in_map[kbase + m0 + r];
    const float* arow = feats + (size_t)in_row * CIN;

    v8f acc0 = {}, acc1 = {}, acc2 = {}, acc3 = {};   // four 16x16 f32 C/D tiles

    #pragma unroll
    for (int c0 = 0; c0 < CIN; c0 += 4) {
        // A 16x4: lane L holds feats[row][c0+klo], feats[row][c0+klo+1]
        v2f a = *(const v2f*)(arow + c0 + klo);
        // B 4x16 per col-block: lane L holds W[c0+klo][col], W[c0+klo+1][col]
        const float* wbase = sW + r * LDS_STRIDE + c0 + klo;
        v2f b0 = *(const v2f*)(wbase);
        v2f b1 = *(const v2f*)(wbase + 16 * LDS_STRIDE);
        v2f b2 = *(const v2f*)(wbase + 32 * LDS_STRIDE);
        v2f b3 = *(const v2f*)(wbase + 48 * LDS_STRIDE);
        acc0 = __builtin_amdgcn_wmma_f32_16x16x4_f32(false, a, false, b0, (short)0, acc0, false, false);
        acc1 = __builtin_amdgcn_wmma_f32_16x16x4_f32(false, a, false, b1, (short)0, acc1, false, false);
        acc2 = __builtin_amdgcn_wmma_f32_16x16x4_f32(false, a, false, b2, (short)0, acc2, false, false);
        acc3 = __builtin_amdgcn_wmma_f32_16x16x4_f32(false, a, false, b3, (short)0, acc3, false, false);
    }

    // Scatter-add: D VGPR v on lane L is output row m0 + v + 8*(L/16), col cb*16 + L%16.
    #pragma unroll
    for (int v = 0; v < 8; ++v) {
        int orow = out_map[kbase + m0 + v + 8 * khalf];
        float* obase = out + (size_t)orow * COUT + r;
        atomicAdd(obase +  0, acc0[v]);
        atomicAdd(obase + 16, acc1[v]);
        atomicAdd(obase + 32, acc2[v]);
        atomicAdd(obase + 48, acc3[v]);
    }
}

extern "C" void kernel_launch(void* const* d_in, const int* in_sizes, int n_in,
                              void* d_out, int out_size, void* d_ws, size_t ws_size,
                              hipStream_t stream) {
    const float* feats   = (const float*)d_in[0];
    const float* weights = (const float*)d_in[1];
    const int*   in_map  = (const int*)d_in[2];
    const int*   out_map = (const int*)d_in[3];
    float*       out     = (float*)d_out;

    const int M = in_sizes[2] / KVOL;     // 50000

    // Scatter target must start at zero (harness poisons d_out).
    zero_f32_kernel<<<2048, 256, 0, stream>>>(out, out_size);

    const int tiles_per_k = M / TILE_M;                                   // 3125
    dim3 grid((tiles_per_k + WAVES_PER_BLOCK - 1) / WAVES_PER_BLOCK,      // 391
              KVOL);                                                      // 27
    spconv_wmma_f32_kernel<<<grid, 256, 0, stream>>>(feats, weights, in_map, out_map, out, M);
}